// SAGE_4733053960618
// MI455X (gfx1250) — compile-verified
//
#include <hip/hip_runtime.h>

typedef __attribute__((ext_vector_type(2))) float v2f;
typedef __attribute__((ext_vector_type(8))) float v8f;

#define NODES 50000
#define EDGES 800000
#define D     128   // feature dim for every layer's input (in=hid=128)

// ---------------- utility: zero a float buffer ----------------
__global__ void zero_f32(float* __restrict__ p, int n) {
    int i = blockIdx.x * blockDim.x + threadIdx.x;
    int stride = gridDim.x * blockDim.x;
    for (; i < n; i += stride) p[i] = 0.0f;
}

// ------------- transpose weights: W[D][ncols] -> Wt[ncols][D] -------------
// Makes the WMMA B-fragment K-pairs (rows kb, kb+1 at one column) contiguous,
// so the GEMM inner loop loads each fragment with a single global_load_b64.
__global__ void transpose_w(const float* __restrict__ W, float* __restrict__ Wt,
                            int ncols) {
    int i = blockIdx.x * blockDim.x + threadIdx.x;   // index into Wt: n*D + k
    int total = ncols * D;
    if (i < total) {
        int n = i / D;
        int k = i - n * D;
        Wt[i] = W[k * ncols + n];
    }
}

// ---------------- degree: deg[dst[e]] += 1 ----------------
__global__ void degree_kernel(const int* __restrict__ dst, float* __restrict__ deg,
                              int n_edges) {
    int i = blockIdx.x * blockDim.x + threadIdx.x;
    int stride = gridDim.x * blockDim.x;
    for (; i < n_edges; i += stride)
        atomicAdd(&deg[dst[i]], 1.0f);
}

// ------------- scatter: agg[dst[e], :] += h[src[e], :] -------------
// One thread handles 4 contiguous features of one edge (float4 gather,
// 4 fp32 global atomics). Threads of a 32-thread group share an edge,
// so src/dst loads broadcast and the gather/atomic addresses coalesce.
__global__ void scatter_kernel(const float* __restrict__ h,
                               const int* __restrict__ src,
                               const int* __restrict__ dst,
                               float* __restrict__ agg, int n_edges) {
    long total  = (long)n_edges * (D / 4);           // 32 chunks per edge
    long i      = (long)blockIdx.x * blockDim.x + threadIdx.x;
    long stride = (long)gridDim.x * blockDim.x;
    for (; i < total; i += stride) {
        int e  = (int)(i >> 5);
        int f4 = ((int)i & 31) << 2;
        int s = src[e], d = dst[e];
        const float4 v = *(const float4*)(h + (long)s * D + f4);
        float* a = agg + (long)d * D + f4;
        atomicAdd(a + 0, v.x);
        atomicAdd(a + 1, v.y);
        atomicAdd(a + 2, v.z);
        atomicAdd(a + 3, v.w);
    }
}

// ------------- fused GEMM: out = h@Wself + (agg/deg)@Wneigh + b [+relu] ------
// Wave32: each wave owns a 32-row x 64-col strip = 2x4 WMMA tiles, fp32 WMMA
// (V_WMMA_F32_16X16X4_F32), K=128 fully unrolled in 32 steps of 4.
// The lane-dependent K sub-offset (half*2) is hoisted into all base pointers,
// so every unrolled load is base + compile-time immediate offset (0..496B):
// no address arithmetic in the hot body, loads free to hoist and clause.
// A frag (32-bit 16x4, ISA 7.12.2): lanes 0-15 = rows M; K pair by lane half.
// B frag (4x16): lanes 0-15 = cols N, same K split; weights pre-transposed to
// [NCOLS][D] so a fragment is one b64 load.  C/D: reg = M off, half = M base.
// Ragged column edge (NCOLS=47): clamp index + zero-select, branch-free.
// Ragged row tail: clamped load rows + predicated epilogue stores.
template <int NCOLS, int TILES, bool COLMASK, bool RELU>
__global__ void __launch_bounds__(32)
sage_gemm_wmma(const float* __restrict__ h,       // [NODES, D]
               const float* __restrict__ agg,     // [NODES, D]
               const float* __restrict__ deg,     // [NODES]
               const float* __restrict__ WselfT,  // [NCOLS, D] (transposed)
               const float* __restrict__ WneighT, // [NCOLS, D] (transposed)
               const float* __restrict__ bias,    // [NCOLS]
               float* __restrict__ out) {         // [NODES, NCOLS]
    const int lane = threadIdx.x;      // 0..31
    const int half = lane >> 4;        // 0/1 -> K sub-offset inside a frag
    const int m    = lane & 15;        // row (A) / col (B,C,D) within a tile
    const int rowb = blockIdx.x * 32;
    const int colb = blockIdx.y * 64;
    const int koff = half * 2;         // lane-dependent K offset, hoisted

    // per-row-block A state (clamped rows for the 50000->50016 tail)
    float inv[2];
    const float* __restrict__ hrow[2];
    const float* __restrict__ grow[2];
#pragma unroll
    for (int rb = 0; rb < 2; ++rb) {
        int r  = rowb + rb * 16 + m;
        int ar = r < NODES ? r : NODES - 1;
        inv[rb]  = 1.0f / fmaxf(deg[ar], 1.0f);
        hrow[rb] = h   + (long)ar * D + koff;
        grow[rb] = agg + (long)ar * D + koff;
    }

    // per-tile column clamp + select, and per-tile transposed-weight row bases
    float sel[TILES];
    const float* __restrict__ wsp[TILES];
    const float* __restrict__ wnp[TILES];
#pragma unroll
    for (int t = 0; t < TILES; ++t) {
        int ncol = colb + t * 16 + m;
        int ncl  = (!COLMASK || ncol < NCOLS) ? ncol : (NCOLS - 1);
        sel[t]   = (!COLMASK || ncol < NCOLS) ? 1.0f : 0.0f;
        wsp[t]   = WselfT  + (long)ncl * D + koff;
        wnp[t]   = WneighT + (long)ncl * D + koff;
    }

    v8f c[2][TILES];
#pragma unroll
    for (int rb = 0; rb < 2; ++rb)
#pragma unroll
        for (int t = 0; t < TILES; ++t)
            c[rb][t] = v8f{0.f, 0.f, 0.f, 0.f, 0.f, 0.f, 0.f, 0.f};

#pragma unroll
    for (int k0 = 0; k0 < D; k0 += 4) {     // fully unrolled: imm offsets only
        v2f a_s[2], a_n[2];
#pragma unroll
        for (int rb = 0; rb < 2; ++rb) {
            const float2 hv = *(const float2*)(hrow[rb] + k0);
            const float2 gv = *(const float2*)(grow[rb] + k0);
            a_s[rb].x = hv.x;
            a_s[rb].y = hv.y;
            a_n[rb].x = gv.x * inv[rb];
            a_n[rb].y = gv.y * inv[rb];
        }
#pragma unroll
        for (int t = 0; t < TILES; ++t) {
            const float2 bs = *(const float2*)(wsp[t] + k0);   // one b64 each
            const float2 bn = *(const float2*)(wnp[t] + k0);
            v2f b_s, b_n;
            b_s.x = bs.x; b_s.y = bs.y;
            b_n.x = bn.x; b_n.y = bn.y;
            if (COLMASK) {                 // zero OOB columns, branch-free
                b_s.x *= sel[t]; b_s.y *= sel[t];
                b_n.x *= sel[t]; b_n.y *= sel[t];
            }
#pragma unroll
            for (int rb = 0; rb < 2; ++rb) {
                c[rb][t] = __builtin_amdgcn_wmma_f32_16x16x4_f32(
                    false, a_s[rb], false, b_s, (short)0, c[rb][t], false, false);
                c[rb][t] = __builtin_amdgcn_wmma_f32_16x16x4_f32(
                    false, a_n[rb], false, b_n, (short)0, c[rb][t], false, false);
            }
        }
    }

#pragma unroll
    for (int rb = 0; rb < 2; ++rb) {
#pragma unroll
        for (int t = 0; t < TILES; ++t) {
            const int ncol = colb + t * 16 + m;
            if (COLMASK && ncol >= NCOLS) continue;
            const float bv = bias[ncol];
#pragma unroll
            for (int r = 0; r < 8; ++r) {
                const int row = rowb + rb * 16 + r + half * 8;
                if (row < NODES) {
                    float v = c[rb][t][r] + bv;
                    if (RELU) v = fmaxf(v, 0.0f);
                    out[(long)row * NCOLS + ncol] = v;
                }
            }
        }
    }
}

extern "C" void kernel_launch(void* const* d_in, const int* in_sizes, int n_in,
                              void* d_out, int out_size, void* d_ws, size_t ws_size,
                              hipStream_t stream) {
    const float* x   = (const float*)d_in[0];
    const int*   src = (const int*)  d_in[1];
    const int*   dst = (const int*)  d_in[2];
    const float* Ws0 = (const float*)d_in[3];
    const float* Wn0 = (const float*)d_in[4];
    const float* b0  = (const float*)d_in[5];
    const float* Ws1 = (const float*)d_in[6];
    const float* Wn1 = (const float*)d_in[7];
    const float* b1  = (const float*)d_in[8];
    const float* Ws2 = (const float*)d_in[9];
    const float* Wn2 = (const float*)d_in[10];
    const float* b2  = (const float*)d_in[11];
    float* out = (float*)d_out;

    // workspace layout (floats):
    //   deg | agg | h1 | h2 | 6 transposed weights   (~77.5 MB)
    float* deg  = (float*)d_ws;
    float* agg  = deg + NODES;
    float* h1   = agg + (long)NODES * D;
    float* h2   = h1  + (long)NODES * D;
    float* wt0s = h2   + (long)NODES * D;
    float* wt0n = wt0s + D * D;
    float* wt1s = wt0n + D * D;
    float* wt1n = wt1s + D * D;
    float* wt2s = wt1n + D * D;
    float* wt2n = wt2s + 47 * D;

    const dim3 blk256(256);
    const dim3 gblk(32);
    const int  rowblks = (NODES + 31) / 32;      // 1563
    const dim3 ggrid128(rowblks, 2);             // 128 cols = 2 x 64-col strips
    const dim3 ggrid47 (rowblks, 1);             // 47 cols, 3 masked tiles

    // one-time: weight transposes + degree (shared by all layers)
    transpose_w<<<dim3((D * D + 255) / 256), blk256, 0, stream>>>(Ws0, wt0s, D);
    transpose_w<<<dim3((D * D + 255) / 256), blk256, 0, stream>>>(Wn0, wt0n, D);
    transpose_w<<<dim3((D * D + 255) / 256), blk256, 0, stream>>>(Ws1, wt1s, D);
    transpose_w<<<dim3((D * D + 255) / 256), blk256, 0, stream>>>(Wn1, wt1n, D);
    transpose_w<<<dim3((47 * D + 255) / 256), blk256, 0, stream>>>(Ws2, wt2s, 47);
    transpose_w<<<dim3((47 * D + 255) / 256), blk256, 0, stream>>>(Wn2, wt2n, 47);
    zero_f32<<<dim3(256), blk256, 0, stream>>>(deg, NODES);
    degree_kernel<<<dim3(2048), blk256, 0, stream>>>(dst, deg, EDGES);

    // ---- layer 0: x -> h1 (relu) ----
    zero_f32<<<dim3(4096), blk256, 0, stream>>>(agg, NODES * D);
    scatter_kernel<<<dim3(8192), blk256, 0, stream>>>(x, src, dst, agg, EDGES);
    sage_gemm_wmma<128, 4, false, true>
        <<<ggrid128, gblk, 0, stream>>>(x, agg, deg, wt0s, wt0n, b0, h1);

    // ---- layer 1: h1 -> h2 (relu) ----
    zero_f32<<<dim3(4096), blk256, 0, stream>>>(agg, NODES * D);
    scatter_kernel<<<dim3(8192), blk256, 0, stream>>>(h1, src, dst, agg, EDGES);
    sage_gemm_wmma<128, 4, false, true>
        <<<ggrid128, gblk, 0, stream>>>(h1, agg, deg, wt1s, wt1n, b1, h2);

    // ---- layer 2: h2 -> out (no relu, 47 classes) ----
    zero_f32<<<dim3(4096), blk256, 0, stream>>>(agg, NODES * D);
    scatter_kernel<<<dim3(8192), blk256, 0, stream>>>(h2, src, dst, agg, EDGES);
    sage_gemm_wmma<47, 3, true, false>
        <<<ggrid47, gblk, 0, stream>>>(h2, agg, deg, wt2s, wt2n, b2, out);
}